// LSKA_79491254714482
// MI455X (gfx1250) — compile-verified
//
#include <hip/hip_runtime.h>

typedef __attribute__((ext_vector_type(2))) float v2f;
typedef __attribute__((ext_vector_type(8))) float v8f;

#define BT 8
#define C  32
#define Dd 32
#define Hh 64
#define Ww 64
#define HW (Hh*Ww)          // 4096
#define S  (Dd*Hh*Ww)       // 131072 (per-channel spatial size)
#define IMG (C*S)           // per-image stride

// ---- kernel 1 tiling ----
#define HT 4                // h rows per tile
#define TH6 (HT+2)          // 6 rows incl. halo
#define TW (Ww+2)           // 66 cols incl. halo
#define DSTRIDE (TH6*TW)    // 396
#define TILE_ELEMS (Dd*TH6*TW)   // 12672
#define INT_H (Dd*HT*TW)         // 8448  (conv-H output region)
#define INT_W (Dd*HT*Ww)         // 8192  (conv-W output region)

// =====================================================================
// Kernel 1: fused depthwise chain
//   convD(3) -> convH(3) -> convW(3) -> 3x dilated convD(5, dil 2)
// One workgroup = one (bt, c, h-tile). Full D and W kept in the tile.
// =====================================================================
__global__ __launch_bounds__(256) void lska_depthwise(
    const float* __restrict__ x,
    const float* __restrict__ w0z, const float* __restrict__ b0z,
    const float* __restrict__ w0x, const float* __restrict__ b0x,
    const float* __restrict__ w0y, const float* __restrict__ b0y,
    const float* __restrict__ wsz, const float* __restrict__ bsz,
    float* __restrict__ hbuf)
{
    extern __shared__ float smem[];
    float* bufA = smem;                 // [Dd][TH6][TW]
    float* bufB = smem + TILE_ELEMS;    // [Dd][TH6][TW]

    const int blk = blockIdx.x;
    const int ht  = blk & 15;           // 16 h-tiles of HT=4
    const int c   = (blk >> 4) & 31;
    const int bt  = blk >> 9;
    const int h0  = ht * HT;
    const int tid = threadIdx.x;

    const float* xc = x + (size_t)bt * IMG + (size_t)c * S;

    const float wz0 = w0z[c*3+0], wz1 = w0z[c*3+1], wz2 = w0z[c*3+2], bz = b0z[c];
    const float wx0 = w0x[c*3+0], wx1 = w0x[c*3+1], wx2 = w0x[c*3+2], bx = b0x[c];
    const float wy0 = w0y[c*3+0], wy1 = w0y[c*3+1], wy2 = w0y[c*3+2], by = b0y[c];
    const float ws0 = wsz[c*5+0], ws1 = wsz[c*5+1], ws2 = wsz[c*5+2],
                ws3 = wsz[c*5+3], ws4 = wsz[c*5+4], bs = bsz[c];

    // ---- load tile (zero outside global H/W range) ----
    for (int i = tid; i < TILE_ELEMS; i += 256) {
        int ww = i % TW;
        int r  = i / TW;
        int hh = r % TH6;
        int d  = r / TH6;
        int gh = h0 + hh - 1;
        int gw = ww - 1;
        float v = 0.0f;
        if ((unsigned)gh < (unsigned)Hh && (unsigned)gw < (unsigned)Ww)
            v = xc[d*HW + gh*Ww + gw];
        bufA[i] = v;
    }
    __syncthreads();

    // ---- stage 1: 3-tap conv along D (zero pad in D), all rows/cols ----
    for (int i = tid; i < TILE_ELEMS; i += 256) {
        int d = i / DSTRIDE;
        float v = bz + wz1 * bufA[i];
        if (d > 0)      v += wz0 * bufA[i - DSTRIDE];
        if (d < Dd - 1) v += wz2 * bufA[i + DSTRIDE];
        bufB[i] = v;
    }
    __syncthreads();

    // ---- stage 2: 3-tap conv along H, interior rows hh=1..HT, all cols ----
    // input rows outside global [0,H) are logically zero
    for (int i = tid; i < INT_H; i += 256) {
        int ww = i % TW;
        int r  = i / TW;
        int hh = (r & 3) + 1;
        int d  = r >> 2;
        int base = d*DSTRIDE + hh*TW + ww;
        int ghm = h0 + hh - 2;   // global index of row hh-1
        int ghp = h0 + hh;       // global index of row hh+1
        float v = bx + wx1 * bufB[base];
        if (ghm >= 0) v += wx0 * bufB[base - TW];
        if (ghp < Hh) v += wx2 * bufB[base + TW];
        bufA[base] = v;
    }
    __syncthreads();

    // ---- stage 3: 3-tap conv along W, interior ww=1..W, rows 1..HT ----
    // cols ww'==0 and ww'==TW-1 map to global -1 / W -> zero
    for (int i = tid; i < INT_W; i += 256) {
        int ww = (i & 63) + 1;
        int r  = i >> 6;
        int hh = (r & 3) + 1;
        int d  = r >> 2;
        int base = d*DSTRIDE + hh*TW + ww;
        float v = by + wy1 * bufA[base];
        if (ww > 1)  v += wy0 * bufA[base - 1];
        if (ww < Ww) v += wy2 * bufA[base + 1];
        bufB[base] = v;
    }
    __syncthreads();

    // ---- stages 4-6: 3x dilated 5-tap conv along D, in registers ----
    // one thread per interior (hh, ww) column; taps at d-4,-2,0,+2,+4
    {
        const int ww = (tid & 63) + 1;
        const int hh = (tid >> 6) + 1;
        float v[Dd], t[Dd];
        #pragma unroll
        for (int d = 0; d < Dd; ++d) v[d] = bufB[d*DSTRIDE + hh*TW + ww];
        #pragma unroll
        for (int p = 0; p < 3; ++p) {
            #pragma unroll
            for (int d = 0; d < Dd; ++d) {
                float a = bs + ws2 * v[d];
                if (d >= 4)      a += ws0 * v[d-4];
                if (d >= 2)      a += ws1 * v[d-2];
                if (d <= Dd - 3) a += ws3 * v[d+2];
                if (d <= Dd - 5) a += ws4 * v[d+4];
                t[d] = a;
            }
            #pragma unroll
            for (int d = 0; d < Dd; ++d) v[d] = t[d];
        }
        const int gh = h0 + hh - 1;
        const int gw = ww - 1;
        float* hc = hbuf + (size_t)bt*IMG + (size_t)c*S + gh*Ww + gw;
        #pragma unroll
        for (int d = 0; d < Dd; ++d) hc[(size_t)d * HW] = v[d];
    }
}

// =====================================================================
// Kernel 2: pointwise 1x1x1 conv as WMMA f32 GEMM + bias + residual mul
//   out[co,s] = x[co,s] * ( sum_ci wp[co,ci]*h[ci,s] + bp[co] )
// Each wave: 64 consecutive spatial positions = 4 N-tiles of 16.
// A layout (16x4 f32): lanes 0-15 = rows M; VGPR0: K=0|K=2, VGPR1: K=1|K=3.
// B layout (4x16 f32): lanes = cols N; VGPR0: K=0|K=2, VGPR1: K=1|K=3.
// C/D layout (16x16 f32): VGPR r -> rows r (lanes 0-15) and r+8 (16-31).
// =====================================================================
__global__ __launch_bounds__(256) void lska_pointwise_wmma(
    const float* h, const float* __restrict__ x,
    const float* __restrict__ wp, const float* __restrict__ bp,
    float* out)
{
    const int lane = threadIdx.x & 31;
    const int wid  = (blockIdx.x * 256 + threadIdx.x) >> 5;  // 0..16383
    const int base = wid * 64;
    const int bt   = base / S;
    const int s    = base - bt * S;

    const float* hb = h   + (size_t)bt*IMG + s;
    const float* xb = x   + (size_t)bt*IMG + s;
    float*       ob = out + (size_t)bt*IMG + s;

    const bool lo   = lane < 16;
    const int  ml   = lane & 15;
    const int  koff = lo ? 0 : 2;   // K offset for this lane half
    const int  roff = lo ? 0 : 8;   // row offset for C/D layout

    // A: wp[co][ci], two M-tiles x 8 K-steps
    v2f a[2][8];
    #pragma unroll
    for (int m = 0; m < 2; ++m) {
        #pragma unroll
        for (int k = 0; k < 8; ++k) {
            const int row = m*16 + ml;
            const int col = 4*k + koff;
            a[m][k].x = wp[row*32 + col];
            a[m][k].y = wp[row*32 + col + 1];
        }
    }

    float bpv[2][8];
    #pragma unroll
    for (int m = 0; m < 2; ++m)
        #pragma unroll
        for (int r = 0; r < 8; ++r)
            bpv[m][r] = bp[m*16 + r + roff];

    #pragma unroll
    for (int t = 0; t < 4; ++t) {
        const int off = t*16 + ml;
        v8f acc0 = {0.f,0.f,0.f,0.f,0.f,0.f,0.f,0.f};
        v8f acc1 = {0.f,0.f,0.f,0.f,0.f,0.f,0.f,0.f};
        #pragma unroll
        for (int k = 0; k < 8; ++k) {
            const int kb = 4*k + koff;
            v2f b;
            b.x = hb[(size_t)kb      * S + off];
            b.y = hb[(size_t)(kb+1)  * S + off];
            acc0 = __builtin_amdgcn_wmma_f32_16x16x4_f32(
                       false, a[0][k], false, b, (short)0, acc0, false, false);
            acc1 = __builtin_amdgcn_wmma_f32_16x16x4_f32(
                       false, a[1][k], false, b, (short)0, acc1, false, false);
        }
        // epilogue: bias + residual multiply + store
        #pragma unroll
        for (int r = 0; r < 8; ++r) {
            const int co0 = r + roff;
            const int co1 = 16 + r + roff;
            const size_t i0 = (size_t)co0 * S + off;
            const size_t i1 = (size_t)co1 * S + off;
            ob[i0] = xb[i0] * (acc0[r] + bpv[0][r]);
            ob[i1] = xb[i1] * (acc1[r] + bpv[1][r]);
        }
    }
}

extern "C" void kernel_launch(void* const* d_in, const int* in_sizes, int n_in,
                              void* d_out, int out_size, void* d_ws, size_t ws_size,
                              hipStream_t stream) {
    const float* x   = (const float*)d_in[0];
    const float* w0z = (const float*)d_in[1];
    const float* b0z = (const float*)d_in[2];
    const float* w0x = (const float*)d_in[3];
    const float* b0x = (const float*)d_in[4];
    const float* w0y = (const float*)d_in[5];
    const float* b0y = (const float*)d_in[6];
    const float* wsz = (const float*)d_in[7];
    const float* bsz = (const float*)d_in[8];
    const float* wp  = (const float*)d_in[9];
    const float* bp  = (const float*)d_in[10];
    float* out = (float*)d_out;

    const size_t hbytes = (size_t)BT * IMG * sizeof(float);
    // Stage h in workspace if it fits, else stage in d_out (safe: kernel 2
    // reads each 16-column tile's h inputs before writing that tile's outputs,
    // and spatial chunks are wave-exclusive).
    float* hbuf = (ws_size >= hbytes) ? (float*)d_ws : out;

    const size_t shmem = (size_t)2 * TILE_ELEMS * sizeof(float);  // ~101 KB
    lska_depthwise<<<BT * C * (Hh / HT), 256, shmem, stream>>>(
        x, w0z, b0z, w0x, b0x, w0y, b0y, wsz, bsz, hbuf);

    lska_pointwise_wmma<<<(BT * S) / (64 * 8), 256, 0, stream>>>(
        hbuf, x, wp, bp, out);
}